// IndexKernelSingle_38216619000009
// MI455X (gfx1250) — compile-verified
//
#include <hip/hip_runtime.h>
#include <hip/hip_bf16.h>

#define THREADS 1024

// Native clang vector types so __builtin_nontemporal_load/store are legal
// (HIP's int4/float4 are class types and are rejected by the builtin).
typedef int   v4i __attribute__((ext_vector_type(4)));
typedef float v4f __attribute__((ext_vector_type(4)));

// Per-pair covariance value: dot(cf[x], cf[y]) + (x==y)*std[x]^2,
// where cf = scf^2 elementwise, so dot = sum_r (scf_x_r * scf_y_r)^2.
__device__ __forceinline__ float cov_pair(const float4* __restrict__ t,
                                          const float* __restrict__ stdv,
                                          int xi, int yi)
{
    float4 a0 = t[2 * xi + 0];
    float4 a1 = t[2 * xi + 1];
    float4 b0 = t[2 * yi + 0];
    float4 b1 = t[2 * yi + 1];

    float p0 = a0.x * b0.x;
    float p1 = a0.y * b0.y;
    float p2 = a0.z * b0.z;
    float p3 = a0.w * b0.w;
    float p4 = a1.x * b1.x;
    float p5 = a1.y * b1.y;
    float p6 = a1.z * b1.z;
    float p7 = a1.w * b1.w;

    float acc = p0 * p0;
    acc = fmaf(p1, p1, acc);
    acc = fmaf(p2, p2, acc);
    acc = fmaf(p3, p3, acc);
    acc = fmaf(p4, p4, acc);
    acc = fmaf(p5, p5, acc);
    acc = fmaf(p6, p6, acc);
    acc = fmaf(p7, p7, acc);

    if (xi == yi) {                // ~1e-4 probability: rare divergence,
        float s = stdv[xi];        // std table (40KB) stays cache-resident
        acc = fmaf(s, s, acc);
    }
    return acc;
}

extern "C" __global__ void __launch_bounds__(THREADS)
covar_gather_kernel(const int*   __restrict__ x,
                    const int*   __restrict__ y,
                    const float* __restrict__ scf,
                    const float* __restrict__ stdv,
                    float*       __restrict__ out,
                    int n_total, int tbl_vec4)
{
    // Dynamic LDS: full sqrt_covar_factor table (C*R floats = 312.5 KB,
    // fits in CDNA5's 320 KB WGP LDS).
    extern __shared__ float4 s_tbl[];

    // ---- Stage table into LDS with gfx1250 async global->LDS DMA ----
    // Dynamic LDS starts right after static LDS (none here -> offset 0).
    const unsigned lds_base = __builtin_amdgcn_groupstaticsize();
    const unsigned tid = threadIdx.x;
    for (unsigned j = tid; j < (unsigned)tbl_vec4; j += THREADS) {
        unsigned ldsa = lds_base + j * 16u;   // LDS byte address (VDST)
        unsigned goff = j * 16u;              // byte offset from SADDR base
        asm volatile("global_load_async_to_lds_b128 %0, %1, %2"
                     :: "v"(ldsa), "v"(goff), "s"(scf)
                     : "memory");
    }
    // Wait for all of this wave's async copies (ASYNCcnt), then workgroup sync.
    asm volatile("s_wait_asynccnt 0" ::: "memory");
    __syncthreads();

    // ---- Streaming gather + rank-8 dot-of-squares ----
    const int gid    = blockIdx.x * THREADS + tid;
    const int stride = gridDim.x * THREADS;
    const int nquad  = n_total >> 2;

    const v4i* __restrict__ x4 = (const v4i*)x;
    const v4i* __restrict__ y4 = (const v4i*)y;
    v4f* __restrict__ out4     = (v4f*)out;

    for (int q = gid; q < nquad; q += stride) {
        // x/y/out are touched exactly once -> non-temporal (TH=NT) so the
        // streams don't evict the hot scf table / std from the caches.
        v4i xi = __builtin_nontemporal_load(&x4[q]);
        v4i yi = __builtin_nontemporal_load(&y4[q]);
        v4f r;
        r.x = cov_pair(s_tbl, stdv, xi.x, yi.x);
        r.y = cov_pair(s_tbl, stdv, xi.y, yi.y);
        r.z = cov_pair(s_tbl, stdv, xi.z, yi.z);
        r.w = cov_pair(s_tbl, stdv, xi.w, yi.w);
        __builtin_nontemporal_store(r, &out4[q]);
    }

    // Scalar tail (N % 4), normally empty for N = 8388608.
    for (int n = (nquad << 2) + gid; n < n_total; n += stride) {
        out[n] = cov_pair(s_tbl, stdv, x[n], y[n]);
    }
}

extern "C" void kernel_launch(void* const* d_in, const int* in_sizes, int n_in,
                              void* d_out, int out_size, void* d_ws, size_t ws_size,
                              hipStream_t stream) {
    const int*   x    = (const int*)d_in[0];
    const int*   y    = (const int*)d_in[1];
    const float* scf  = (const float*)d_in[2];   // [C, R] fp32
    const float* stdv = (const float*)d_in[3];   // [C]    fp32
    float*       out  = (float*)d_out;

    const int N          = in_sizes[0];
    const int tbl_floats = in_sizes[2];          // C*R = 80000
    const int tbl_vec4   = tbl_floats / 4;       // 20000 x 16B = 320000 B

    const size_t smem = (size_t)tbl_floats * sizeof(float);

    // ~2 quads per thread: 8.4M elems / 4 / (1024 thr) -> 1024 blocks.
    int nquad  = N >> 2;
    int blocks = (nquad + THREADS * 2 - 1) / (THREADS * 2);
    if (blocks < 1) blocks = 1;
    if (blocks > 8192) blocks = 8192;

    covar_gather_kernel<<<blocks, THREADS, smem, stream>>>(
        x, y, scf, stdv, out, N, tbl_vec4);
}